// MCPBRNN_Generic_NOscaling_10196252360842
// MI455X (gfx1250) — compile-verified
//
#include <hip/hip_runtime.h>
#include <math.h>

#define SPIN_N 10000
#define TRAIN_N 800000

typedef __attribute__((ext_vector_type(2))) float v2f;
typedef __attribute__((ext_vector_type(8))) float v8f;

#ifndef __has_builtin
#define __has_builtin(x) 0
#endif

// gfx1250 has V_TANH_F32 (listed as a TRANS op in the CDNA5 ISA). Prefer the
// clang builtin when declared; otherwise emit the instruction directly.
#if __has_builtin(__builtin_amdgcn_tanhf)
#define TANHF(x) __builtin_amdgcn_tanhf(x)
#elif __has_builtin(__builtin_amdgcn_tanh_f32)
#define TANHF(x) __builtin_amdgcn_tanh_f32(x)
#else
__device__ __forceinline__ float tanhf_asm(float x) {
  float r;
  asm("v_tanh_f32 %0, %1" : "=v"(r) : "v"(x));
  return r;
}
#define TANHF(x) tanhf_asm(x)
#endif

// One sigmoid gate, pre-folded through the tanh identity:
//   G*sigmoid(b + w*v) = Ah + Ah*tanh(k1*v + k0),  k1=.5w, k0=.5b, Ah=.5G
struct GateK { float k1, k0, Ah; };

__device__ __forceinline__ GateK make_gate(float w, float b, float G) {
  GateK g;
  g.k1 = 0.5f * w; g.k0 = 0.5f * b; g.Ah = 0.5f * G;
  return g;
}

__device__ __forceinline__ float gate_val(const GateK& g, float v) {
  float T = TANHF(g.k1 * v + g.k0);
  return g.Ah * T + g.Ah;
}

// One serial step. u1 and the precomputed leak gate ol arrive as a packed
// pair. Critical chain: v_fma -> v_tanh_f32 -> v_fma (3 dependent ops).
__device__ __forceinline__ float scan_step(float c, float u1, float ol,
                                           const GateK& go) {
  float To   = TANHF(go.k1 * c + go.k0);   // chain
  float a    = go.Ah * c;                  // parallel
  float base = (c + u1) - ol * c - a;      // parallel (add, fma, sub)
  return base - a * To;                    // chain (fma)
}

// ---------------------------------------------------------------------------
// Kernel A: the serial recurrence. Reads packed (u1, ol) pairs prepared by
// the prep kernel, writes ONLY c_n. Unrolled x16 with software prefetch.
// ---------------------------------------------------------------------------
__global__ void mcp_scan_kernel(
    const float* __restrict__ pk,   // packed pairs: pk[2t]=u1_t, pk[2t+1]=ol_t
    const float* __restrict__ wr_yom, const float* __restrict__ wr_ylm,
    const float* __restrict__ wr_yfm,
    const float* __restrict__ b0_yom, const float* __restrict__ w1_yom,
    float* __restrict__ c_n, int B) {
  if (blockIdx.x != 0 || threadIdx.x != 0) return;

  const float e_o = __expf(wr_yom[0]);
  const float e_l = __expf(wr_ylm[0]);
  const float e_f = __expf(wr_yfm[0]);
  const GateK go = make_gate(w1_yom[0], b0_yom[0], e_o / (e_o + e_l + e_f));

  float c = 0.0f;
  int t = 0;
  const int Bmain = B & ~15;
  for (; t < Bmain; t += 16) {
    __builtin_prefetch(pk + 2 * (size_t)t + 512, 0, 0);  // 2 KB ahead
#pragma unroll
    for (int j = 0; j < 16; ++j) {
      const int i = t + j;
      v2f u = *(const v2f*)(pk + 2 * (size_t)i);
      c_n[i] = c;                        // pre-update cell state
      c = scan_step(c, u.x, u.y, go);
    }
  }
  for (; t < B; ++t) {
    v2f u = *(const v2f*)(pk + 2 * (size_t)t);
    c_n[t] = c;
    c = scan_step(c, u.x, u.y, go);
  }
}

// ---------------------------------------------------------------------------
// Kernel B: parallel prep. Packs (u1_t, ol_t) pairs into the h_nout output
// slot (2B contiguous floats, fully overwritten later by the post kernel).
// ---------------------------------------------------------------------------
__global__ __launch_bounds__(256) void mcp_prep_kernel(
    const float* __restrict__ x,
    const float* __restrict__ wr_yom, const float* __restrict__ wr_ylm,
    const float* __restrict__ wr_yfm,
    const float* __restrict__ b0_ylm, const float* __restrict__ w2_ylm,
    float* __restrict__ pk, int B) {
  const float e_o = __expf(wr_yom[0]);
  const float e_l = __expf(wr_ylm[0]);
  const float e_f = __expf(wr_yfm[0]);
  const GateK gl = make_gate(w2_ylm[0], b0_ylm[0], e_l / (e_o + e_l + e_f));

  for (int b = blockIdx.x * blockDim.x + threadIdx.x; b < B;
       b += gridDim.x * blockDim.x) {
    v2f u = *(const v2f*)(x + 2 * (size_t)b);
    v2f p; p.x = u.x; p.y = gate_val(gl, u.y);
    *(v2f*)(pk + 2 * (size_t)b) = p;
  }
}

// ---------------------------------------------------------------------------
// Kernel C: per-wave (sum, sumsq) partials over y_obs[SPIN:TRAIN] on the
// matrix pipe (V_WMMA_F32_16X16X4_F32 with B = ones -> row-sum accumulate).
// ---------------------------------------------------------------------------
__global__ __launch_bounds__(256) void mcp_reduce_kernel(
    const float* __restrict__ y, float* __restrict__ partials,
    int n, int numWaves) {
  const int lane = threadIdx.x & 31;
  const int wave = (int)((blockIdx.x * blockDim.x + threadIdx.x) >> 5);
  const float* __restrict__ ys = y + SPIN_N;

  v8f accS = {};
  v8f accQ = {};
  v2f ones; ones.x = 1.0f; ones.y = 1.0f;

  const int fullChunks = n >> 6;
  for (int chunk = wave; chunk < fullChunks; chunk += numWaves) {
    v2f a = *(const v2f*)(ys + ((size_t)chunk << 6) + (lane << 1));
    v2f q; q.x = a.x * a.x; q.y = a.y * a.y;
#if __has_builtin(__builtin_amdgcn_wmma_f32_16x16x4_f32)
    accS = __builtin_amdgcn_wmma_f32_16x16x4_f32(false, a, false, ones,
                                                 (short)0, accS, false, false);
    accQ = __builtin_amdgcn_wmma_f32_16x16x4_f32(false, q, false, ones,
                                                 (short)0, accQ, false, false);
#else
    accS[0] += a.x + a.y;
    accQ[0] += q.x + q.y;
#endif
  }

  float ls = 0.0f, lq = 0.0f;
#pragma unroll
  for (int r = 0; r < 8; ++r) { ls += accS[r]; lq += accQ[r]; }

#if __has_builtin(__builtin_amdgcn_wmma_f32_16x16x4_f32)
  // D columns are replicated; rows 0..7 in lane 0, rows 8..15 in lane 16.
  float s  = __shfl(ls, 0, 32) + __shfl(ls, 16, 32);
  float qq = __shfl(lq, 0, 32) + __shfl(lq, 16, 32);
#else
  float s = ls, qq = lq;
  for (int off = 16; off > 0; off >>= 1) {
    s  += __shfl_xor(s,  off, 32);
    qq += __shfl_xor(qq, off, 32);
  }
#endif

  float ts = 0.0f, tq = 0.0f;
  if (wave == 0) {
    int i0 = (fullChunks << 6) + (lane << 1);
    if (i0 < n)     { float v = ys[i0];     ts += v; tq += v * v; }
    if (i0 + 1 < n) { float v = ys[i0 + 1]; ts += v; tq += v * v; }
  }
  for (int off = 16; off > 0; off >>= 1) {
    ts += __shfl_xor(ts, off, 32);
    tq += __shfl_xor(tq, off, 32);
  }

  if (lane == 0) {
    partials[2 * wave]     = s + ts;
    partials[2 * wave + 1] = qq + tq;
  }
}

// ---------------------------------------------------------------------------
// Kernel D: deterministic double-precision combine -> obsstd (ddof=1).
// ---------------------------------------------------------------------------
__global__ void mcp_finalize_kernel(const float* __restrict__ partials,
                                    float* __restrict__ std_out,
                                    int numWaves, int n) {
  if (blockIdx.x == 0 && threadIdx.x == 0) {
    double s = 0.0, q = 0.0;
    for (int i = 0; i < numWaves; ++i) {
      s += (double)partials[2 * i];
      q += (double)partials[2 * i + 1];
    }
    double mean = s / (double)n;
    double var  = (q - s * mean) / (double)(n - 1);
    std_out[0] = (float)sqrt(var > 0.0 ? var : 0.0);
  }
}

// ---------------------------------------------------------------------------
// Kernel E: fused parallel epilogue. From c_n and x, recompute all remaining
// outputs: h_n, l_n, zeros, Gate_oo, Gate_ol, Gate_f, h_nout (both columns,
// overwriting the prep scratch), obs_std fills.
// ---------------------------------------------------------------------------
__global__ __launch_bounds__(256) void mcp_post_kernel(
    const float* __restrict__ x,
    const float* __restrict__ wr_yom, const float* __restrict__ wr_ylm,
    const float* __restrict__ wr_yfm,
    const float* __restrict__ b0_yom, const float* __restrict__ w1_yom,
    const float* __restrict__ b0_ylm, const float* __restrict__ w2_ylm,
    const float* __restrict__ std_in,
    float* __restrict__ out, int B) {
  const float e_o = __expf(wr_yom[0]);
  const float e_l = __expf(wr_ylm[0]);
  const float e_f = __expf(wr_yfm[0]);
  const float denom = e_o + e_l + e_f;
  const GateK go = make_gate(w1_yom[0], b0_yom[0], e_o / denom);
  const GateK gl = make_gate(w2_ylm[0], b0_ylm[0], e_l / denom);
  const float sd = std_in[0];

  float* __restrict__ h_n       = out;
  const float* __restrict__ c_n = out + (size_t)B;
  float* __restrict__ l_n       = out + 2 * (size_t)B;
  float* __restrict__ z1        = out + 3 * (size_t)B;
  float* __restrict__ z2        = out + 4 * (size_t)B;
  float* __restrict__ g_oo      = out + 5 * (size_t)B;
  float* __restrict__ g_ol      = out + 6 * (size_t)B;
  float* __restrict__ g_f       = out + 7 * (size_t)B;
  float* __restrict__ h_nout    = out + 8 * (size_t)B;
  float* __restrict__ o_std     = out + 10 * (size_t)B;

  for (int b = blockIdx.x * blockDim.x + threadIdx.x; b < B;
       b += gridDim.x * blockDim.x) {
    float cp = c_n[b];
    float u2 = x[2 * (size_t)b + 1];
    float oo = gate_val(go, cp);
    float ol = gate_val(gl, u2);
    float f  = 1.0f - oo - ol;
    float h  = oo * cp;
    h_n[b]  = h;
    l_n[b]  = ol * cp;
    z1[b]   = 0.0f;
    z2[b]   = 0.0f;
    g_oo[b] = oo;
    g_ol[b] = ol;
    g_f[b]  = f;
    h_nout[2 * (size_t)b]     = h;
    h_nout[2 * (size_t)b + 1] = sd;
    o_std[b] = sd;
  }
}

// ---------------------------------------------------------------------------
extern "C" void kernel_launch(void* const* d_in, const int* in_sizes, int n_in,
                              void* d_out, int out_size, void* d_ws,
                              size_t ws_size, hipStream_t stream) {
  const float* x       = (const float*)d_in[0];   // (B,1,2)
  const float* y_obs   = (const float*)d_in[1];   // (B,1)
  const float* wr_yom  = (const float*)d_in[2];
  const float* wr_ylm  = (const float*)d_in[3];
  const float* wr_yfm  = (const float*)d_in[4];
  const float* b0_yom  = (const float*)d_in[5];
  const float* w1_yom  = (const float*)d_in[6];
  const float* b0_ylm  = (const float*)d_in[7];
  const float* w2_ylm  = (const float*)d_in[8];
  // d_in[9] = epoch, d_in[10] = time_lag: unused by the math.

  float* out = (float*)d_out;
  const int B = in_sizes[1];            // 1,000,000
  const int n = TRAIN_N - SPIN_N;       // 790,000

  const int RED_BLOCKS = 64, RED_THREADS = 256;
  const int numWaves = RED_BLOCKS * (RED_THREADS / 32);  // 512

  float* partials = (float*)d_ws;                 // 2*numWaves floats
  float* std_out  = partials + 2 * numWaves;      // 1 float
  float* c_n      = out + (size_t)B;              // output slot 2
  float* pk       = out + 8 * (size_t)B;          // h_nout slot, reused as
                                                  // (u1, ol) scratch

  mcp_prep_kernel<<<512, 256, 0, stream>>>(x, wr_yom, wr_ylm, wr_yfm, b0_ylm,
                                           w2_ylm, pk, B);
  mcp_reduce_kernel<<<RED_BLOCKS, RED_THREADS, 0, stream>>>(y_obs, partials, n,
                                                            numWaves);
  mcp_finalize_kernel<<<1, 32, 0, stream>>>(partials, std_out, numWaves, n);
  mcp_scan_kernel<<<1, 32, 0, stream>>>(pk, wr_yom, wr_ylm, wr_yfm, b0_yom,
                                        w1_yom, c_n, B);
  mcp_post_kernel<<<512, 256, 0, stream>>>(x, wr_yom, wr_ylm, wr_yfm, b0_yom,
                                           w1_yom, b0_ylm, w2_ylm, std_out,
                                           out, B);
}